// CL_MLP_88175678587650
// MI455X (gfx1250) — compile-verified
//
#include <hip/hip_runtime.h>
#include <hip/hip_bf16.h>
#include <math.h>

// ---------------------------------------------------------------------------
// MI455X (gfx1250) implementation.
// Compute-bound fp32 workload -> V_WMMA_F32_16X16X4_F32 (full precision,
// matrix pipe). ~103 GFLOP total vs ~136MB traffic (5.8us @ 23.3TB/s).
// GEMM W-tiles staged via CDNA5 async global->LDS copies (ASYNCcnt),
// double-buffered, shared by all 8 waves of the block.
// ---------------------------------------------------------------------------

typedef __attribute__((ext_vector_type(2))) float v2f;
typedef __attribute__((ext_vector_type(8))) float v8f;

#define HID    1024
#define T_LEN  1024
#define BATCH  8
#define ROWS   (T_LEN * BATCH)        // 8192 rows per tensor
#define NPAIR  2046                   // n = 2*(t-1)
#define NREP   4092                   // 2n
#define NREP_PAD 4096
#define HSECT  (BATCH * T_LEN * HID)  // 8388608 floats per output tensor
#define INV_TEMP 10.0f
#define WPAD   72                     // padded LDS row stride (bank-conflict free)

__device__ __forceinline__ v8f wmma_f32x4(v2f a, v2f b, v8f c) {
  // D = A(16x4,f32) * B(4x16,f32) + C(16x16,f32)
  return __builtin_amdgcn_wmma_f32_16x16x4_f32(
      /*neg_a=*/false, a, /*neg_b=*/false, b,
      /*c_mod=*/(short)0, c, /*reuse_a=*/false, /*reuse_b=*/false);
}

// Map rep-row index i in [0,4092) to its row in d_out.
// reps = concat( h1[7,1:], h2[7,1:], h1[7,:-1], h2[7,:-1] )
__device__ __forceinline__ const float* rep_ptr(const float* out, int i) {
  int which, t;
  if (i < 1023)      { which = 0; t = i + 1; }
  else if (i < 2046) { which = 1; t = i - 1022; }
  else if (i < 3069) { which = 0; t = i - 2046; }
  else               { which = 1; t = i - 3069; }
  return out + (size_t)which * HSECT + ((size_t)7 * T_LEN + t) * HID;
}

// ---------------------------------------------------------------------------
// GEMM: Y = relu(X @ W + bias).  Block = 8 waves -> 128(M) x 64(N) tile;
// one wave -> 16(M) x 64(N).  W tile (16K x 64N) is block-shared: staged into
// LDS with global_load_async_to_lds_b128 (double buffered, ASYNCcnt).
// REMAP=true: input rows come from [T,B,H] layout (layer 1 transpose fold-in).
// ---------------------------------------------------------------------------
template <bool REMAP>
__global__ void __launch_bounds__(256)
mlp_gemm(const float* __restrict__ X1, const float* __restrict__ X2,
         const float* __restrict__ W,  const float* __restrict__ bias,
         float* __restrict__ O1, float* __restrict__ O2) {
  __shared__ float wbuf[2][16][WPAD];   // 2 x 16 k-rows x 64 cols (+pad)

  const int tid  = threadIdx.x;
  const int lane = tid & 31;
  const int wave = tid >> 5;
  const int half = lane >> 4;     // 0|1
  const int ln   = lane & 15;
  const int kb   = half << 1;     // K sub-offset 0|2 (A/B frag layout)
  const int m0   = blockIdx.x * 128 + wave * 16;
  const int n0   = blockIdx.y * 64;

  const float* X = blockIdx.z ? X2 : X1;
  float*       O = blockIdx.z ? O2 : O1;

  const int rowA = m0 + ln;
  const float* aptr;
  if (REMAP) {  // row r of [B,T,H] view: b=r>>10, t=r&1023 ; X is [T,B,H]
    aptr = X + (size_t)(rowA & (T_LEN - 1)) * (BATCH * HID)
             + (size_t)(rowA >> 10) * HID + kb;
  } else {
    aptr = X + (size_t)rowA * HID + kb;
  }

  // Staging: thread t copies W[k0 + t/16][n0 + (t%16)*4 .. +3] (16B) to LDS.
  const int sr = tid >> 4;               // 0..15 : k-row within stage
  const int sc = (tid & 15) << 2;        // 0,4,..,60 : col within stage
  const float* wsrc = W + (size_t)sr * HID + n0 + sc;

  auto stage = [&](int buf, int k0) {
    unsigned lds = (unsigned)(uintptr_t)&wbuf[buf][sr][sc];
    unsigned long long ga =
        (unsigned long long)(uintptr_t)(wsrc + (size_t)k0 * HID);
    asm volatile("global_load_async_to_lds_b128 %0, %1, off"
                 :: "v"(lds), "v"(ga) : "memory");
  };

  stage(0, 0);

  v8f acc0 = {}, acc1 = {}, acc2 = {}, acc3 = {};
  for (int k0 = 0; k0 < HID; k0 += 16) {
    const int cur = (k0 >> 4) & 1;
    asm volatile("s_wait_asynccnt 0x0" ::: "memory");
    __syncthreads();                       // cur buffer ready for all waves
    if (k0 + 16 < HID) stage(cur ^ 1, k0 + 16);  // overlap next fill

#pragma unroll
    for (int ks = 0; ks < 16; ks += 4) {
      v2f a = *(const v2f*)(aptr + k0 + ks);     // A[m, k..k+1] (per frag half)
      const float* w0 = &wbuf[cur][ks + kb][ln]; // B[k][n], k rows WPAD apart
      v2f b0 = { w0[0],       w0[WPAD]      };
      v2f b1 = { w0[16],      w0[WPAD + 16] };
      v2f b2 = { w0[32],      w0[WPAD + 32] };
      v2f b3 = { w0[48],      w0[WPAD + 48] };
      acc0 = wmma_f32x4(a, b0, acc0);
      acc1 = wmma_f32x4(a, b1, acc1);
      acc2 = wmma_f32x4(a, b2, acc2);
      acc3 = wmma_f32x4(a, b3, acc3);
    }
  }

  const int mrow = m0 + half * 8;   // D layout: lanes>=16 hold M=8..15
#pragma unroll
  for (int s = 0; s < 4; ++s) {
    v8f acc = (s == 0) ? acc0 : (s == 1) ? acc1 : (s == 2) ? acc2 : acc3;
    const int n = n0 + s * 16 + ln;
    const float bv = bias[n];
#pragma unroll
    for (int v = 0; v < 8; ++v) {
      float val = acc[v] + bv;
      val = val > 0.0f ? val : 0.0f;  // relu
      O[(size_t)(mrow + v) * HID + n] = val;
    }
  }
}

// ---------------------------------------------------------------------------
// Per-row inverse norms of the 4092 rep rows; zero-init denom/pairlog.
// ---------------------------------------------------------------------------
__global__ void __launch_bounds__(256)
norm_kernel(const float* __restrict__ out, float* __restrict__ rnorm,
            float* __restrict__ denom, float* __restrict__ pairlog) {
  const int lane = threadIdx.x & 31;
  const int i = blockIdx.x * 8 + (threadIdx.x >> 5);
  if (i >= NREP_PAD) return;
  if (i >= NREP) {
    if (lane == 0) { rnorm[i] = 0.f; denom[i] = 0.f; pairlog[i] = 0.f; }
    return;
  }
  const float* p = rep_ptr(out, i);
  float s = 0.f;
  for (int k = lane; k < HID; k += 32) { float x = p[k]; s += x * x; }
#pragma unroll
  for (int off = 16; off; off >>= 1) s += __shfl_xor(s, off, 32);
  if (lane == 0) {
    rnorm[i]   = 1.0f / fmaxf(sqrtf(s), 1e-12f);  // torch normalize eps
    denom[i]   = 0.f;
    pairlog[i] = 0.f;
  }
}

// ---------------------------------------------------------------------------
// sim = (Z Z^T) with normalization folded in: logit = dot * ri * rj / TEMP.
// One wave: fixed 16-row i-tile, 32 j-tiles; register-accumulated row sums.
// ---------------------------------------------------------------------------
__global__ void __launch_bounds__(256)
sim_kernel(const float* __restrict__ out, const float* __restrict__ rnorm,
           float* __restrict__ denom, float* __restrict__ pairlog) {
  const int lane = threadIdx.x & 31;
  const int wave = threadIdx.x >> 5;
  const int half = lane >> 4;
  const int ln   = lane & 15;
  const int kb   = half << 1;
  const int i0   = (blockIdx.x * 8 + wave) * 16;

  int ia = i0 + ln; if (ia > NREP - 1) ia = NREP - 1;   // clamp pad rows
  const float* aptr = rep_ptr(out, ia) + kb;

  float ri[8];
#pragma unroll
  for (int v = 0; v < 8; ++v) {
    int gi = i0 + half * 8 + v;
    ri[v] = rnorm[gi > NREP - 1 ? NREP - 1 : gi];
  }

  float rowsum[8];
#pragma unroll
  for (int v = 0; v < 8; ++v) rowsum[v] = 0.f;

  for (int jj = 0; jj < 32; ++jj) {
    const int j0 = (blockIdx.y * 32 + jj) * 16;
    int jb = j0 + ln; if (jb > NREP - 1) jb = NREP - 1;
    const float* bptr = rep_ptr(out, jb) + kb;   // B col n = rep row j

    v8f acc = {};
    for (int kk = 0; kk < HID; kk += 4) {
      v2f a = *(const v2f*)(aptr + kk);
      v2f b = *(const v2f*)(bptr + kk);
      acc = wmma_f32x4(a, b, acc);
    }

    const int gj = j0 + ln;
    const float rj = rnorm[gj > NREP - 1 ? NREP - 1 : gj];
#pragma unroll
    for (int v = 0; v < 8; ++v) {
      const int gi = i0 + half * 8 + v;
      if (gi < NREP && gj < NREP) {
        const float logit = acc[v] * ri[v] * rj * INV_TEMP;
        if (gi != gj) rowsum[v] += expf(logit);   // denom excludes diagonal
        const int pair = (gi < NPAIR) ? gi + NPAIR : gi - NPAIR;
        if (gj == pair) pairlog[gi] = logit;      // unique writer
      }
    }
  }

  // reduce each row sum across its 16-lane half, then one atomic per row
#pragma unroll
  for (int v = 0; v < 8; ++v) {
    float s = rowsum[v];
    s += __shfl_xor(s, 1, 32);
    s += __shfl_xor(s, 2, 32);
    s += __shfl_xor(s, 4, 32);
    s += __shfl_xor(s, 8, 32);
    if (ln == 0) {
      const int gi = i0 + half * 8 + v;
      if (gi < NREP) atomicAdd(&denom[gi], s);
    }
  }
}

// ---------------------------------------------------------------------------
// loss = ( sum_i log(denom_i) - sum_i pairlogit_i ) / (2n)
// ---------------------------------------------------------------------------
__global__ void __launch_bounds__(256)
loss_kernel(const float* __restrict__ denom, const float* __restrict__ pairlog,
            float* __restrict__ loss_out) {
  __shared__ float sm[256];
  float s = 0.f;
  for (int i = threadIdx.x; i < NREP; i += 256)
    s += logf(denom[i]) - pairlog[i];
  sm[threadIdx.x] = s;
  __syncthreads();
  for (int off = 128; off; off >>= 1) {
    if (threadIdx.x < off) sm[threadIdx.x] += sm[threadIdx.x + off];
    __syncthreads();
  }
  if (threadIdx.x == 0) loss_out[0] = sm[0] / (float)NREP;
}

// ---------------------------------------------------------------------------
extern "C" void kernel_launch(void* const* d_in, const int* in_sizes, int n_in,
                              void* d_out, int out_size, void* d_ws, size_t ws_size,
                              hipStream_t stream) {
  const float* h1 = (const float*)d_in[0];
  const float* h2 = (const float*)d_in[1];
  const float* W1 = (const float*)d_in[2];
  const float* b1 = (const float*)d_in[3];
  const float* W2 = (const float*)d_in[4];
  const float* b2 = (const float*)d_in[5];

  float* out = (float*)d_out;                 // [h1 | h2 | loss]
  float* Y       = (float*)d_ws;              // layer-1 acts: 2*8192*1024 f32
  float* rnorm   = Y + (size_t)2 * ROWS * HID;
  float* denom   = rnorm + NREP_PAD;
  float* pairlog = denom + NREP_PAD;

  const dim3 blk(256);
  // Layer 1 (with [T,B,H]->[B,T,H] row remap) -> workspace
  mlp_gemm<true ><<<dim3(ROWS / 128, HID / 64, 2), blk, 0, stream>>>(
      h1, h2, W1, b1, Y, Y + (size_t)ROWS * HID);
  // Layer 2 -> d_out sections
  mlp_gemm<false><<<dim3(ROWS / 128, HID / 64, 2), blk, 0, stream>>>(
      Y, Y + (size_t)ROWS * HID, W2, b2, out, out + (size_t)HSECT);
  // Row norms + scratch init
  norm_kernel<<<dim3(NREP_PAD / 8), blk, 0, stream>>>(out, rnorm, denom, pairlog);
  // Similarity / softmax denominators via WMMA
  sim_kernel<<<dim3(32, 8), blk, 0, stream>>>(out, rnorm, denom, pairlog);
  // Final scalar
  loss_kernel<<<1, blk, 0, stream>>>(denom, pairlog, out + (size_t)2 * HSECT);
}